// CustomLoss_37306085933679
// MI455X (gfx1250) — compile-verified
//
#include <hip/hip_runtime.h>
#include <hip/hip_bf16.h>
#include <math.h>

// Focal loss over [N=2097152, C=128] fp32 logits.
// Memory-bound: 1 GiB streamed once -> coalesced b128 NT loads, WMMA row-sum.

typedef __attribute__((ext_vector_type(2))) float v2f;
typedef __attribute__((ext_vector_type(4))) float v4f;
typedef __attribute__((ext_vector_type(8))) float v8f;

#define C_DIM      128
#define ROW_STRIDE 132               // 128 + 4 pad floats -> conflict-free LDS reads
#define WAVES_PB   4                 // 128-thread blocks (wave32 -> 4 waves)
#define TILE_ROWS  16                // one WMMA M-tile per wave

__global__ __launch_bounds__(WAVES_PB * 32) void
focal_main_kernel(const float* __restrict__ outputs,
                  const int* __restrict__ targets,
                  float* __restrict__ partials,
                  int nRows)
{
    __shared__ float lds[WAVES_PB * TILE_ROWS * ROW_STRIDE];
    __shared__ float wsum[WAVES_PB];

    const int lid   = threadIdx.x & 31;          // lane in wave32
    const int wv    = threadIdx.x >> 5;          // wave in block
    const int gwave = blockIdx.x * WAVES_PB + wv;
    const int nWav  = gridDim.x * WAVES_PB;
    const int woff  = wv * TILE_ROWS * ROW_STRIDE;
    const int tiles = nRows / TILE_ROWS;

    const v2f ones = (v2f){1.0f, 1.0f};          // B matrix: all-ones 4x16
    float ceAcc = 0.0f;

    for (int tile = gwave; tile < tiles; tile += nWav) {
        const int baseRow = tile * TILE_ROWS;
        const float* tilePtr = outputs + (size_t)baseRow * C_DIM;

        // Hint-prefetch this wave's next tile (global_prefetch_b8).
        if (tile + nWav < tiles) {
            const float* np = outputs + (size_t)(tile + nWav) * C_DIM * TILE_ROWS;
            __builtin_prefetch(np + lid * 64, 0, 1);
        }

        // ---- Stage 16 rows: coalesced NT b128 loads, on-the-fly row max ----
        float rm = 0.0f;                         // lane r (r<16) keeps max(row r)
#pragma unroll
        for (int r = 0; r < TILE_ROWS; ++r) {
            const v4f* rp = (const v4f*)(tilePtr + r * C_DIM) + lid;
            v4f v = __builtin_nontemporal_load(rp);    // 32 lanes x 16B = one row
            float m = fmaxf(fmaxf(v.x, v.y), fmaxf(v.z, v.w));
#pragma unroll
            for (int off = 16; off >= 1; off >>= 1)
                m = fmaxf(m, __shfl_xor(m, off, 32));
            if (lid == r) rm = m;
            *((v4f*)(&lds[woff + r * ROW_STRIDE]) + lid) = v;  // ds_store_b128
        }
        __builtin_amdgcn_wave_barrier();         // order LDS stores vs. loads

        // Max of the row this lane feeds into the WMMA A operand (row = lid&15).
        const float rmAll = __shfl(rm, lid & 15, 32);

        // ---- Row-sum of exp via V_WMMA_F32_16X16X4_F32, accumulate K=128 ----
        // A layout (16x4 fp32): lanes 0-15 -> (M=lane, K=0/1), lanes 16-31 -> (M=lane-16, K=2/3)
        v8f acc = (v8f){0.f, 0.f, 0.f, 0.f, 0.f, 0.f, 0.f, 0.f};
        const int arow = lid & 15;
        const int ksub = (lid >> 4) << 1;        // 0 for low half, 2 for high half
        const float* aBase = &lds[woff + arow * ROW_STRIDE + ksub];
#pragma unroll
        for (int k0 = 0; k0 < C_DIM; k0 += 4) {
            v2f x = *(const v2f*)(aBase + k0);   // ds_load_b64, conflict-free
            v2f a;
            a.x = __expf(x.x - rmAll);
            a.y = __expf(x.y - rmAll);
            acc = __builtin_amdgcn_wmma_f32_16x16x4_f32(
                false, a, false, ones, (short)0, acc, false, false);
        }
        // acc[j]: lanes 0-15 hold rowsum(M=j), lanes 16-31 hold rowsum(M=j+8)

        // ---- Gather logit[target] and accumulate CE (lanes 0-7 & 16-23) ----
        float s = acc[0];
        s = ((lid & 7) == 1) ? (float)acc[1] : s;
        s = ((lid & 7) == 2) ? (float)acc[2] : s;
        s = ((lid & 7) == 3) ? (float)acc[3] : s;
        s = ((lid & 7) == 4) ? (float)acc[4] : s;
        s = ((lid & 7) == 5) ? (float)acc[5] : s;
        s = ((lid & 7) == 6) ? (float)acc[6] : s;
        s = ((lid & 7) == 7) ? (float)acc[7] : s;

        const int  row    = (lid < 16) ? lid : (lid - 8);   // lanes 16-23 -> rows 8-15
        const bool active = (lid & 15) < 8;
        const float gm = __shfl(rm, row & 15, 32);          // that row's max
        if (active) {
            const int t  = targets[baseRow + row];
            const float xt = lds[woff + row * ROW_STRIDE + t];
            ceAcc += __logf(s) + gm - xt;                   // -log_softmax[t]
        }
        __builtin_amdgcn_wave_barrier();         // LDS reuse next tile
    }

    // ---- Deterministic reduction: wave tree -> block -> per-block partial ----
#pragma unroll
    for (int off = 16; off >= 1; off >>= 1)
        ceAcc += __shfl_xor(ceAcc, off, 32);
    if (lid == 0) wsum[wv] = ceAcc;
    __syncthreads();
    if (threadIdx.x == 0) {
        float b = 0.0f;
#pragma unroll
        for (int w = 0; w < WAVES_PB; ++w) b += wsum[w];
        partials[blockIdx.x] = b;
    }
}

__global__ __launch_bounds__(256) void
focal_finish_kernel(const float* __restrict__ partials, int n, float* __restrict__ out)
{
    __shared__ float sm[256];
    float s = 0.0f;
    for (int i = threadIdx.x; i < n; i += 256) s += partials[i];  // fixed order
    sm[threadIdx.x] = s;
    __syncthreads();
    for (int k = 128; k > 0; k >>= 1) {
        if (threadIdx.x < k) sm[threadIdx.x] += sm[threadIdx.x + k];
        __syncthreads();
    }
    if (threadIdx.x == 0) {
        const float ce = sm[0];
        const float pt = __expf(-ce);
        const float om = 1.0f - pt;
        out[0] = om * om * ce;   // GAMMA = 2
    }
}

extern "C" void kernel_launch(void* const* d_in, const int* in_sizes, int n_in,
                              void* d_out, int out_size, void* d_ws, size_t ws_size,
                              hipStream_t stream)
{
    const float* outputs = (const float*)d_in[0];
    const int*   targets = (const int*)d_in[1];
    float*       out     = (float*)d_out;
    float*       parts   = (float*)d_ws;

    const int N = in_sizes[1];          // 2,097,152 rows (divisible by 16)

    constexpr int BLOCKS  = 4096;       // 16 KB of partials in d_ws
    constexpr int THREADS = WAVES_PB * 32;

    focal_main_kernel<<<BLOCKS, THREADS, 0, stream>>>(outputs, targets, parts, N);
    focal_finish_kernel<<<1, 256, 0, stream>>>(parts, BLOCKS, out);
}